// CenterLoss_76845554860177
// MI455X (gfx1250) — compile-verified
//
#include <hip/hip_runtime.h>

typedef __attribute__((ext_vector_type(2))) float v2f;
typedef __attribute__((ext_vector_type(8))) float v8f;

#define BATCH     256
#define FEAT_DIM  512
#define NUM_CLS   100000

// One wave (32 threads) per 16-row group. Computes diag(E * E^T) with
// V_WMMA_F32_16X16X4_F32 where E[i][k] = x[row_i][k] - centers[label_i][k].
// diag(D)[i] = squared L2 distance of row i to its own class center.
__global__ __launch_bounds__(32) void center_dist_wmma_kernel(
    const float* __restrict__ x,
    const int*   __restrict__ labels,
    const float* __restrict__ centers,
    float*       __restrict__ row_dist)
{
    const int lane  = threadIdx.x;        // 0..31, one full wave
    const int group = blockIdx.x;         // 0..15 (256 rows / 16)
    const int r     = lane & 15;          // matrix row index M (== N) held by this lane
    const int row   = group * 16 + r;
    const int khalf = (lane < 16) ? 0 : 2; // A: lanes 0-15 hold K={0,1}, lanes 16-31 K={2,3}

    const int lab = labels[row];
    const float* __restrict__ xr = x       + (size_t)row * FEAT_DIM;
    const float* __restrict__ cr = centers + (size_t)lab * FEAT_DIM;

    v8f acc = {};
    #pragma unroll 4
    for (int k0 = 0; k0 < FEAT_DIM; k0 += 4) {
        const int k = k0 + khalf;
        const float2 xv = *(const float2*)(xr + k);
        const float2 cv = *(const float2*)(cr + k);
        v2f e;
        e.x = xv.x - cv.x;
        e.y = xv.y - cv.y;
        // Gram matrix is symmetric: per-lane A and B operand contents coincide,
        // so pass `e` for both. D += A(16x4) * B(4x16), f32 end-to-end.
        acc = __builtin_amdgcn_wmma_f32_16x16x4_f32(
            /*neg_a=*/false, e, /*neg_b=*/false, e,
            /*c_mod=*/(short)0, acc, /*reuse_a=*/false, /*reuse_b=*/false);
    }

    // Diagonal of 16x16 f32 D: row i at (lane=i, vgpr=i) for i<8,
    // (lane=i+16, vgpr=i-8) for i>=8  ->  acc[lane & 7] on lanes 0-7 and 24-31.
    const int idx = lane & 7;
    float d = acc[0];
    #pragma unroll
    for (int i = 1; i < 8; ++i)
        if (idx == i) d = acc[i];

    if (lane < 8 || lane >= 24) {
        const int rg = lane & 15;  // 0..7 (lanes 0-7) or 8..15 (lanes 24-31)
        d = fminf(fmaxf(d, 1e-12f), 1e12f);   // clip(dist, 1e-12, 1e12)
        row_dist[group * 16 + rg] = d;
    }
}

// Deterministic fixed-order reduction of the 256 per-row distances.
// Adds the clip-floor contribution of the (C-1) masked-out columns per row.
__global__ __launch_bounds__(32) void center_loss_reduce_kernel(
    const float* __restrict__ row_dist,
    float*       __restrict__ out)
{
    if (threadIdx.x == 0) {
        float s = 0.0f;
        for (int i = 0; i < BATCH; ++i)
            s += row_dist[i];
        out[0] = s / (float)BATCH + (float)(NUM_CLS - 1) * 1e-12f;
    }
}

extern "C" void kernel_launch(void* const* d_in, const int* in_sizes, int n_in,
                              void* d_out, int out_size, void* d_ws, size_t ws_size,
                              hipStream_t stream) {
    (void)in_sizes; (void)n_in; (void)out_size; (void)ws_size;
    const float* x       = (const float*)d_in[0];
    const int*   labels  = (const int*)  d_in[1];
    const float* centers = (const float*)d_in[2];
    float*       out     = (float*)d_out;
    float*       row_d   = (float*)d_ws;   // 256 floats of scratch

    center_dist_wmma_kernel<<<BATCH / 16, 32, 0, stream>>>(x, labels, centers, row_d);
    center_loss_reduce_kernel<<<1, 32, 0, stream>>>(row_d, out);
}